// TEAT_GT_26276609917381
// MI455X (gfx1250) — compile-verified
//
#include <hip/hip_runtime.h>
#include <math.h>

#define NB     60000
#define NPAIR  50000
#define F_IMG  1000
#define NCLS   37
#define D_FEAT 2048
#define D_REP  968
#define D_EMB  200
#define D_TOK  1168
#define K_TOT  4096          // 2 * D_FEAT (subj half | obj half)
#define M_PAD  1024          // 32 supertiles * 32 rows
#define N_PAD  992           // 31 supertiles * 32 rows
#define ST_M   32            // supertile grid (each wave: 32x32 output)
#define ST_N   31

typedef __attribute__((ext_vector_type(2))) float v2f;
typedef __attribute__((ext_vector_type(8))) float v8f;

// ---------------------------------------------------------------- workspace
static constexpr size_t align256(size_t x) { return (x + 255) & ~size_t(255); }
static constexpr size_t OFF_AGG = 0;                                             // M_PAD x K_TOT f32
static constexpr size_t OFF_WT  = align256(OFF_AGG + (size_t)M_PAD * K_TOT * 4); // N_PAD x K_TOT f32
static constexpr size_t OFF_GO  = align256(OFF_WT  + (size_t)N_PAD * K_TOT * 4); // NPAIR f32 (obj gates -> weights)
static constexpr size_t OFF_GP  = align256(OFF_GO  + (size_t)NPAIR * 4);         // F f32 (person gate -> weight)
static constexpr size_t OFF_SS  = align256(OFF_GP  + (size_t)F_IMG * 4);         // F f32 (sum of subj weights)
static constexpr size_t OFF_SO  = align256(OFF_SS  + (size_t)F_IMG * 4);         // F f32 (sum of obj weights)
static constexpr size_t OFF_VS  = align256(OFF_SO  + (size_t)F_IMG * 4);         // D_FEAT f32 (W_subj @ gw)
static constexpr size_t OFF_VO  = align256(OFF_VS  + (size_t)D_FEAT * 4);        // D_FEAT f32 (W_obj @ gw)
static constexpr size_t OFF_EG  = align256(OFF_VO  + (size_t)D_FEAT * 4);        // NCLS f32 (embed @ gw_emb)
static constexpr size_t OFF_CC  = align256(OFF_EG  + (size_t)NCLS * 4);          // 2 f32 (bias dot + gate_b)
static constexpr size_t OFF_PI  = align256(OFF_CC  + 2 * 4);                     // F i32 person_idx (segment starts)
static constexpr size_t OFF_SR  = align256(OFF_PI  + (size_t)F_IMG * 4);         // F i32 subj_rows

// ---------------------------------------------------------------- K1: prep
__global__ void k_prep(const int* __restrict__ im_idx, const int* __restrict__ pair_idx,
                       const float* __restrict__ embed, const float* __restrict__ gate_w,
                       const float* __restrict__ b_subj, const float* __restrict__ b_obj,
                       const float* __restrict__ gate_b,
                       int* person_idx, int* subj_rows, float* egate, float* cconst) {
  int tid = threadIdx.x;
  for (int f = tid; f < F_IMG; f += 256) {
    int lo = 0, hi = NPAIR;                 // lower_bound(im_idx, f)
    while (lo < hi) { int mid = (lo + hi) >> 1; if (im_idx[mid] < f) lo = mid + 1; else hi = mid; }
    person_idx[f] = lo;
    subj_rows[f]  = pair_idx[2 * lo + 0];
  }
  for (int c = tid; c < NCLS; c += 256) {
    float s = 0.f;
    for (int k = 0; k < D_EMB; ++k) s += embed[c * D_EMB + k] * gate_w[D_REP + k];
    egate[c] = s;
  }
  if (tid == 0) {
    float cs = gate_b[0], co = gate_b[0];
    for (int j = 0; j < D_REP; ++j) { cs += b_subj[j] * gate_w[j]; co += b_obj[j] * gate_w[j]; }
    cconst[0] = cs; cconst[1] = co;
  }
}

// ---------------------------------------------------- K2: v = W @ gate_w[:D_REP]
__global__ void k_gvec(const float* __restrict__ W_subj, const float* __restrict__ W_obj,
                       const float* __restrict__ gate_w, float* vsub, float* vobj) {
  int gtid = blockIdx.x * 256 + threadIdx.x;
  int wave = gtid >> 5, lane = gtid & 31;
  if (wave >= 2 * D_FEAT) return;
  int d = wave >> 1;
  const float* W = (wave & 1) ? W_obj : W_subj;
  float s = 0.f;
  for (int j = lane; j < D_REP; j += 32) s += W[(size_t)d * D_REP + j] * gate_w[j];
  for (int o = 16; o > 0; o >>= 1) s += __shfl_down(s, o, 32);
  if (lane == 0) ((wave & 1) ? vobj : vsub)[d] = s;
}

// ------------------------------------------- K3: Wt[n][k] = [W_subj;W_obj][k][n]
__global__ void k_wt(const float* __restrict__ W_subj, const float* __restrict__ W_obj,
                     float* __restrict__ Wt) {
  size_t i = (size_t)blockIdx.x * 256 + threadIdx.x;
  if (i >= (size_t)N_PAD * K_TOT) return;
  int n = (int)(i / K_TOT), k = (int)(i % K_TOT);
  float v = 0.f;
  if (n < D_REP) v = (k < D_FEAT) ? W_subj[(size_t)k * D_REP + n]
                                  : W_obj[(size_t)(k - D_FEAT) * D_REP + n];
  Wt[i] = v;
}

// ------------------------------------------- K4: one wave per token -> gate
__global__ void k_gate(const float* __restrict__ features, const int* __restrict__ pair_idx,
                       const int* __restrict__ labels, const int* __restrict__ subj_rows,
                       const float* __restrict__ vsub, const float* __restrict__ vobj,
                       const float* __restrict__ egate, const float* __restrict__ cconst,
                       float* gate_p, float* gate_o) {
  int gtid = blockIdx.x * 256 + threadIdx.x;
  int wave = gtid >> 5, lane = gtid & 31;
  if (wave >= F_IMG + NPAIR) return;
  int row; const float* v; float cadd; float* outp;
  if (wave < F_IMG) { row = subj_rows[wave]; v = vsub; cadd = cconst[0]; outp = gate_p + wave; }
  else { int p = wave - F_IMG; row = pair_idx[2 * p + 1]; v = vobj; cadd = cconst[1]; outp = gate_o + p; }
  const float4* fr = (const float4*)(features + (size_t)row * D_FEAT);
  const float4* vv = (const float4*)v;
  float s = 0.f;
  for (int i = lane; i < D_FEAT / 4; i += 32) {
    float4 a = fr[i], b = vv[i];
    s += a.x * b.x + a.y * b.y + a.z * b.z + a.w * b.w;
  }
  for (int o = 16; o > 0; o >>= 1) s += __shfl_down(s, o, 32);
  if (lane == 0) *outp = s + egate[labels[row]] + cadd;
}

// ---------------------- K5: per-segment softmax; overwrite gates with weights
__global__ void k_soft(const int* __restrict__ person_idx,
                       float* gate_p, float* gate_o, float* ssub, float* sobj) {
  __shared__ float red[256];
  int f = blockIdx.x, tid = threadIdx.x;
  int s = person_idx[f];
  int e = (f == F_IMG - 1) ? NPAIR : person_idx[f + 1];
  float gp = gate_p[f];
  float mx = (tid == 0) ? gp : -3.4e38f;
  for (int p = s + tid; p < e; p += 256) mx = fmaxf(mx, gate_o[p]);
  red[tid] = mx; __syncthreads();
  for (int o = 128; o > 0; o >>= 1) { if (tid < o) red[tid] = fmaxf(red[tid], red[tid + o]); __syncthreads(); }
  float m = red[0]; __syncthreads();
  float z = (tid == 0) ? expf(gp - m) : 0.f;
  for (int p = s + tid; p < e; p += 256) z += expf(gate_o[p] - m);
  red[tid] = z; __syncthreads();
  for (int o = 128; o > 0; o >>= 1) { if (tid < o) red[tid] += red[tid + o]; __syncthreads(); }
  float inv = 1.0f / red[0];
  for (int p = s + tid; p < e; p += 256) gate_o[p] = expf(gate_o[p] - m) * inv;
  if (tid == 0) { float wp = expf(gp - m) * inv; gate_p[f] = wp; ssub[f] = wp; sobj[f] = 1.0f - wp; }
}

// -------- K6: Agg[f] = [ wp*feat_subj | sum_obj w*feat ]; emb part of output
__global__ void k_acc(const float* __restrict__ features, const int* __restrict__ pair_idx,
                      const int* __restrict__ labels, const int* __restrict__ subj_rows,
                      const int* __restrict__ person_idx, const float* __restrict__ embed,
                      const float* __restrict__ gate_p, const float* __restrict__ gate_o,
                      float* __restrict__ Agg, float* __restrict__ out) {
  int f = blockIdx.x, tid = threadIdx.x;
  float* arow = Agg + (size_t)f * K_TOT;
  if (f >= F_IMG) { for (int c = tid; c < K_TOT; c += 256) arow[c] = 0.f; return; }
  int s = person_idx[f];
  int e = (f == F_IMG - 1) ? NPAIR : person_idx[f + 1];
  float wp = gate_p[f];
  int srow = subj_rows[f];
  int c0 = tid * 8;                              // 256 threads * 8 = 2048 cols
  float acc[8];
  #pragma unroll
  for (int j = 0; j < 8; ++j) acc[j] = wp * features[(size_t)srow * D_FEAT + c0 + j];
  #pragma unroll
  for (int j = 0; j < 8; ++j) arow[c0 + j] = acc[j];
  #pragma unroll
  for (int j = 0; j < 8; ++j) acc[j] = 0.f;
  for (int p = s; p < e; ++p) {
    float w = gate_o[p];
    int row = pair_idx[2 * p + 1];
    const float* fr = features + (size_t)row * D_FEAT + c0;
    if (p + 1 < e) __builtin_prefetch(features + (size_t)pair_idx[2 * (p + 1) + 1] * D_FEAT + c0, 0, 0);
    #pragma unroll
    for (int j = 0; j < 8; ++j) acc[j] += w * fr[j];
  }
  #pragma unroll
  for (int j = 0; j < 8; ++j) arow[D_FEAT + c0 + j] = acc[j];
  if (tid < D_EMB) {
    float a = wp * embed[labels[srow] * D_EMB + tid];
    for (int p = s; p < e; ++p)
      a += gate_o[p] * embed[labels[pair_idx[2 * p + 1]] * D_EMB + tid];
    out[(size_t)f * D_TOK + D_REP + tid] = a;
  }
}

// --- K7: WMMA GEMM, 32x32 supertile per wave (2x2 register blocking, 1 load/WMMA)
__global__ void k_gemm(const float* __restrict__ Agg, const float* __restrict__ Wt,
                       const float* __restrict__ b_subj, const float* __restrict__ b_obj,
                       const float* __restrict__ ssub, const float* __restrict__ sobj,
                       float* __restrict__ out) {
  int tid = threadIdx.x;
  int st  = blockIdx.x * 8 + (tid >> 5);         // one 32x32 supertile per wave
  if (st >= ST_M * ST_N) return;                 // wave-uniform exit (EXEC all-1 for WMMA)
  int lane = tid & 31, half = lane >> 4, l15 = lane & 15;
  int m0 = (st / ST_N) * 32;
  int n0 = (st % ST_N) * 32;
  // A fragment (16x4 f32): lane m holds {K=2h, K=2h+1}; B via n-major Wt: same pattern.
  const float* ap0 = Agg + (size_t)(m0 + l15) * K_TOT + 2 * half;
  const float* ap1 = ap0 + (size_t)16 * K_TOT;
  const float* bp0 = Wt  + (size_t)(n0 + l15) * K_TOT + 2 * half;
  const float* bp1 = bp0 + (size_t)16 * K_TOT;
  v8f acc00 = {0.f,0.f,0.f,0.f,0.f,0.f,0.f,0.f};
  v8f acc01 = acc00, acc10 = acc00, acc11 = acc00;
  #pragma unroll 2
  for (int kb = 0; kb < K_TOT; kb += 4) {
    v2f a0 = *(const v2f*)(ap0 + kb);
    v2f a1 = *(const v2f*)(ap1 + kb);
    v2f b0 = *(const v2f*)(bp0 + kb);
    v2f b1 = *(const v2f*)(bp1 + kb);
    acc00 = __builtin_amdgcn_wmma_f32_16x16x4_f32(false, a0, false, b0, (short)0, acc00, false, false);
    acc01 = __builtin_amdgcn_wmma_f32_16x16x4_f32(false, a0, false, b1, (short)0, acc01, false, false);
    acc10 = __builtin_amdgcn_wmma_f32_16x16x4_f32(false, a1, false, b0, (short)0, acc10, false, false);
    acc11 = __builtin_amdgcn_wmma_f32_16x16x4_f32(false, a1, false, b1, (short)0, acc11, false, false);
  }
  v8f accs[2][2] = {{acc00, acc01}, {acc10, acc11}};
  #pragma unroll
  for (int j = 0; j < 2; ++j) {
    int n = n0 + j * 16 + l15;
    if (n < D_REP) {
      float bs = b_subj[n], bo = b_obj[n];
      #pragma unroll
      for (int i = 0; i < 2; ++i) {
        #pragma unroll
        for (int r = 0; r < 8; ++r) {
          int m = m0 + i * 16 + r + half * 8;    // C/D layout: VGPR r -> M=r (lanes<16) / r+8
          if (m < F_IMG) out[(size_t)m * D_TOK + n] = accs[i][j][r] + ssub[m] * bs + sobj[m] * bo;
        }
      }
    }
  }
}

// ---------------------------------------------------------------- launcher
extern "C" void kernel_launch(void* const* d_in, const int* in_sizes, int n_in,
                              void* d_out, int out_size, void* d_ws, size_t ws_size,
                              hipStream_t stream) {
  const float* features = (const float*)d_in[0];
  // d_in[1] = boxes: dead code for the returned value
  const int*   labels   = (const int*)d_in[2];
  const int*   im_idx   = (const int*)d_in[3];
  const int*   pair_idx = (const int*)d_in[4];
  const float* embed    = (const float*)d_in[5];
  const float* W_subj   = (const float*)d_in[6];
  const float* b_subj   = (const float*)d_in[7];
  const float* W_obj    = (const float*)d_in[8];
  const float* b_obj    = (const float*)d_in[9];
  const float* gate_w   = (const float*)d_in[10];
  const float* gate_b   = (const float*)d_in[11];
  float* out = (float*)d_out;

  char* ws = (char*)d_ws;
  float* Agg   = (float*)(ws + OFF_AGG);
  float* Wt    = (float*)(ws + OFF_WT);
  float* g_o   = (float*)(ws + OFF_GO);
  float* g_p   = (float*)(ws + OFF_GP);
  float* ssub  = (float*)(ws + OFF_SS);
  float* sobj  = (float*)(ws + OFF_SO);
  float* vsub  = (float*)(ws + OFF_VS);
  float* vobj  = (float*)(ws + OFF_VO);
  float* egate = (float*)(ws + OFF_EG);
  float* cc    = (float*)(ws + OFF_CC);
  int*   pidx  = (int*)(ws + OFF_PI);
  int*   srows = (int*)(ws + OFF_SR);

  k_prep<<<1, 256, 0, stream>>>(im_idx, pair_idx, embed, gate_w, b_subj, b_obj, gate_b,
                                pidx, srows, egate, cc);
  k_gvec<<<(2 * D_FEAT) / 8, 256, 0, stream>>>(W_subj, W_obj, gate_w, vsub, vobj);
  k_wt<<<(int)(((size_t)N_PAD * K_TOT + 255) / 256), 256, 0, stream>>>(W_subj, W_obj, Wt);
  k_gate<<<(F_IMG + NPAIR + 7) / 8, 256, 0, stream>>>(features, pair_idx, labels, srows,
                                                      vsub, vobj, egate, cc, g_p, g_o);
  k_soft<<<F_IMG, 256, 0, stream>>>(pidx, g_p, g_o, ssub, sobj);
  k_acc<<<M_PAD, 256, 0, stream>>>(features, pair_idx, labels, srows, pidx, embed,
                                   g_p, g_o, Agg, out);
  k_gemm<<<(ST_M * ST_N + 7) / 8, 256, 0, stream>>>(Agg, Wt, b_subj, b_obj,
                                                    ssub, sobj, out);
}